// CrossPatchModule_48120813584595
// MI455X (gfx1250) — compile-verified
//
#include <hip/hip_runtime.h>

typedef unsigned int  u32;
typedef unsigned long long u64;
typedef __attribute__((ext_vector_type(4))) u32   v4u;
typedef __attribute__((ext_vector_type(8))) int   v8i;
typedef __attribute__((ext_vector_type(4))) int   v4i;
typedef __attribute__((ext_vector_type(4))) float v4f;

// Reference collapses to a patch-roll + per-tile scalar bias:
//   B=4, C=64, H=W=512, PH=PW=8, PN=64, kh=kw=64
//   out[b,c, ph2*64+r, pw2*64+s] = x[b,c, ph*64+r, pw*64+s] + abs_pos[c*64+m]
//     j = ph2*8+pw2 (output tile), m = (c+j)&63, ph = m>>3, pw = m&7
// Pure data movement (537 MB @ 23.3 TB/s ~= 23 us floor). Loads via the CDNA5
// Tensor Data Mover (one 2D D# per 64x64 tile), bias-add from LDS, NT 128-bit stores.

__global__ __launch_bounds__(256) void crosspatch_tdm(
    const float* __restrict__ x,
    const float* __restrict__ abs_pos,
    float* __restrict__ out)
{
    __shared__ __align__(16) float tile[64 * 64];   // 16 KB staging tile

    const u32 blk = blockIdx.x;          // (b*64 + c)*64 + j
    const u32 j   = blk & 63u;
    const u32 c   = (blk >> 6) & 63u;
    const u32 b   = blk >> 12;

    const u32 m   = (c + j) & 63u;
    const u32 ph  = m >> 3,  pw  = m & 7u;
    const u32 ph2 = j >> 3,  pw2 = j & 7u;

    const u64 img = (u64)(b * 64u + c) * (512ull * 512ull);
    const float* in_base  = x   + img + (u64)(ph  * 64u) * 512u + pw  * 64u;
    float*       out_base = out + img + (u64)(ph2 * 64u) * 512u + pw2 * 64u;

    const float bias = abs_pos[c * 64u + m];   // one scalar per tile

    // --- TDM: DMA the 64x64 f32 tile (row stride 512 floats) into LDS.
    // TDM issue ignores EXEC, so gate with a *scalar* per-wave branch:
    if (__builtin_amdgcn_readfirstlane((int)threadIdx.x) == 0) {   // wave 0 only
        const u64 ga = (u64)(uintptr_t)in_base;
        const u32 la = (u32)(uintptr_t)&tile[0];  // low 32 bits of flat addr == LDS byte offset

        v4u g0;
        g0.x = 1u;                                   // count=1 valid descriptor; no gather
        g0.y = la;                                   // lds_addr
        g0.z = (u32)ga;                              // global_addr[31:0]
        g0.w = (u32)((ga >> 32) & 0x01FFFFFFu)       // global_addr[56:32]
             | (2u << 30);                           // type = 2 ("image")

        v8i g1;
        g1[0] = (int)(2u << 16);    // workgroup_mask=0, data_size=2 (4 B), no pad/iterate
        g1[1] = (int)(64u << 16);   // atomic_barrier_addr=0; tensor_dim0 = 64 (lo16)
        g1[2] = (int)(64u << 16);   // tensor_dim0 hi = 0;   tensor_dim1 = 64 (lo16)
        g1[3] = (int)(64u << 16);   // tensor_dim1 hi = 0;   tile_dim0   = 64
        g1[4] = (int)(64u);         // tile_dim1 = 64;       tile_dim2   = 0 (2D)
        g1[5] = (int)(512u);        // tensor_dim0_stride = 512 elements (lo32)
        g1[6] = 0;                  // stride hi16 = 0; tensor_dim1_stride unused (2D)
        g1[7] = 0;

        v4i gz4 = {0, 0, 0, 0};                 // groups 2/3 unused for 2D tensors
        v8i gz8 = {0, 0, 0, 0, 0, 0, 0, 0};     // trailing group unused
        __builtin_amdgcn_tensor_load_to_lds(g0, g1, gz4, gz4, gz8, 0);
        __builtin_amdgcn_s_wait_tensorcnt(0);
    }
    // Escape the LDS array (TDM only sees its address as an integer in the D#),
    // so the compiler cannot fold/delete the never-stored shared loads.
    asm volatile("" : : "v"(&tile[0]) : "memory");
    __syncthreads();

    // --- bias-add + non-temporal 128-bit streaming stores (output never re-read) ---
    const v4f* sp = (const v4f*)tile;
    #pragma unroll
    for (u32 k = 0; k < 4; ++k) {
        const u32 q  = threadIdx.x + k * 256u;  // float4 index within tile [0,1024)
        const u32 r  = q >> 4;                  // row within tile
        const u32 c4 = q & 15u;                 // float4 column within row
        v4f v = sp[q];
        v.x += bias; v.y += bias; v.z += bias; v.w += bias;
        __builtin_nontemporal_store(v, (v4f*)(out_base + (u64)r * 512u + c4 * 4u));
    }
}

extern "C" void kernel_launch(void* const* d_in, const int* in_sizes, int n_in,
                              void* d_out, int out_size, void* d_ws, size_t ws_size,
                              hipStream_t stream) {
    (void)in_sizes; (void)n_in; (void)out_size; (void)d_ws; (void)ws_size;
    const float* x  = (const float*)d_in[0];   // (4,64,512,512) f32
    const float* ap = (const float*)d_in[1];   // (1,1,64,64,1,1) f32 -> flat [c*64+m]
    float* out = (float*)d_out;                // (4,64,512,512) f32
    crosspatch_tdm<<<dim3(4 * 64 * 64), dim3(256), 0, stream>>>(x, ap, out);
}